// LorenzOscillator_65343632441879
// MI455X (gfx1250) — compile-verified
//
#include <hip/hip_runtime.h>
#include <hip/hip_bf16.h>

typedef __attribute__((ext_vector_type(2))) float v2f;
typedef __attribute__((ext_vector_type(8))) float v8f;

#define SIGMA 10.0f
#define RHO   28.0f
#define BETA  (8.0f / 3.0f)
#define HSTEP 0.01f

#define NB   32
#define NS   8000
#define NHID 256
#define NROWS (NB * NS)          // 256000
#define NTILES (NROWS / 16)      // 16000
#define WAVES_PER_BLOCK 8

// ---------------------------------------------------------------------------
// Kernel 1: serial RK4 scan. One wave32; lane b integrates batch b's chain.
// coup[b,s,d] = x[b,s]*A[d] + C[d]  (IN_DIM==1 collapses both einsums).
// States written as float4 rows {x,y,z,1} at r = b*NS + s. The 1.0 pad is
// the bias channel: the GEMM's K=4 WMMA multiplies it against b_out[h] so
// the bias add comes out of the matrix unit with C = inline 0.
// ---------------------------------------------------------------------------
__device__ __forceinline__ void lorenz_deriv(float sx, float sy, float sz,
                                             float c0, float c1, float c2,
                                             float& dx, float& dy, float& dz) {
    dx = SIGMA * (sy - sx) + c0;
    dy = sx * (RHO - sz) - sy + c1;
    dz = sx * sy - BETA * sz + c2;
}

__global__ void lorenz_scan(const float* __restrict__ x,      // [NB*NS]
                            const float* __restrict__ W_in,   // [3]
                            const float* __restrict__ b_in,   // [3]
                            const float* __restrict__ W_coup, // [3*3] (d,c)
                            float4* __restrict__ st)          // [NROWS]
{
    const int b = threadIdx.x;   // launched with exactly 32 threads
    if (b >= NB) return;

    float A0 = 0.f, A1 = 0.f, A2 = 0.f;
    float C0 = 0.f, C1 = 0.f, C2 = 0.f;
#pragma unroll
    for (int c = 0; c < 3; ++c) {
        const float wi = W_in[c];
        const float bi = b_in[c];
        A0 += W_coup[0 * 3 + c] * wi;  C0 += W_coup[0 * 3 + c] * bi;
        A1 += W_coup[1 * 3 + c] * wi;  C1 += W_coup[1 * 3 + c] * bi;
        A2 += W_coup[2 * 3 + c] * wi;  C2 += W_coup[2 * 3 + c] * bi;
    }

    float sx = 0.f, sy = 0.f, sz = 0.f;
    const float* xb = x + (size_t)b * NS;
    float4* stb = st + (size_t)b * NS;

    for (int s = 0; s < NS; ++s) {
        const float xv = xb[s];
        const float c0 = fmaf(xv, A0, C0);
        const float c1 = fmaf(xv, A1, C1);
        const float c2 = fmaf(xv, A2, C2);

        float k1x, k1y, k1z, k2x, k2y, k2z, k3x, k3y, k3z, k4x, k4y, k4z;
        lorenz_deriv(sx, sy, sz, c0, c1, c2, k1x, k1y, k1z);
        lorenz_deriv(sx + 0.5f * HSTEP * k1x, sy + 0.5f * HSTEP * k1y,
                     sz + 0.5f * HSTEP * k1z, c0, c1, c2, k2x, k2y, k2z);
        lorenz_deriv(sx + 0.5f * HSTEP * k2x, sy + 0.5f * HSTEP * k2y,
                     sz + 0.5f * HSTEP * k2z, c0, c1, c2, k3x, k3y, k3z);
        lorenz_deriv(sx + HSTEP * k3x, sy + HSTEP * k3y,
                     sz + HSTEP * k3z, c0, c1, c2, k4x, k4y, k4z);

        const float w = HSTEP / 6.0f;
        sx += w * (k1x + 2.f * k2x + 2.f * k3x + k4x);
        sy += w * (k1y + 2.f * k2y + 2.f * k3y + k4y);
        sz += w * (k1z + 2.f * k2z + 2.f * k3z + k4z);

        stb[s] = make_float4(sx, sy, sz, 1.0f);   // pad = bias channel
    }
}

// ---------------------------------------------------------------------------
// Kernel 2: out[r,h] = states[r,0:3] . W_out[h,0:3] + b_out[h]
// (256000 x 4) x (4 x 256) via V_WMMA_F32_16X16X4_F32, C = inline 0.
//
// Bias is folded into K: A's 4th channel is 1.0 (from the scan), B's 4th
// row is b_out[h]. Branch-free operand feeds:
//   A: raw b64 at st + r*16 + (lane>=16 ? 8 : 0) -> {c0,c1} / {c2,1}
//   B: W_out packed [256][4] = {w0,w1,w2,bias} in LDS; b64 at
//      h*16 + (lane>=16 ? 8 : 0) -> {w0,w1} / {w2,bias}
// N-loop fully unrolled: 16 WMMAs per wave. Output stored non-temporally
// (write-once 262 MB stream; don't rinse L2).
// ---------------------------------------------------------------------------
__global__ void lorenz_out_gemm(const float* __restrict__ st,   // [NROWS*4] floats
                                const float* __restrict__ W_out,// [NHID*3]
                                const float* __restrict__ b_out,// [NHID]
                                float* __restrict__ out)        // [NROWS*NHID]
{
    __shared__ float wpad[NHID * 4];   // {W0,W1,W2,bias} per h

    // Block-cooperative fill (256 threads, one h each).
    {
        const int h = threadIdx.x;     // blockDim.x == 256
        wpad[h * 4 + 0] = W_out[h * 3 + 0];
        wpad[h * 4 + 1] = W_out[h * 3 + 1];
        wpad[h * 4 + 2] = W_out[h * 3 + 2];
        wpad[h * 4 + 3] = b_out[h];
    }
    __syncthreads();

    const int lane = threadIdx.x & 31;
    const int wave = threadIdx.x >> 5;
    const int tile = blockIdx.x * WAVES_PER_BLOCK + wave;  // < NTILES

    const int m     = lane & 15;           // row (A) / col (B,D) within tile
    const int hiofs = (lane >> 4) << 3;    // 0 or 8 bytes: K pair {0,1}/{2,3}

    // A operand: exact 2-float slice of the padded state row.
    const int r = tile * 16 + m;
    const v2f a = *reinterpret_cast<const v2f*>(
        reinterpret_cast<const char*>(st) + (size_t)r * 16 + hiofs);

    const int rowbase = tile * 16 + ((lane >> 4) << 3);   // D rows M=v (+8 hi)

#pragma unroll
    for (int nt = 0; nt < NHID / 16; ++nt) {
        const int h = nt * 16 + m;

        // B operand: exact 2-float slice of the packed weight row (LDS).
        const v2f bm = *reinterpret_cast<const v2f*>(
            reinterpret_cast<const char*>(wpad) + h * 16 + hiofs);

        const v8f c = {};   // inline 0 accumulator; bias rides in K=3

        // D = A x B + 0   ->  v_wmma_f32_16x16x4_f32 v[D], v[A], v[B], 0
        v8f d = __builtin_amdgcn_wmma_f32_16x16x4_f32(
            /*neg_a=*/false, a, /*neg_b=*/false, bm,
            /*c_mod=*/(short)0, c, /*reuse_a=*/false, /*reuse_b=*/false);

#pragma unroll
        for (int v = 0; v < 8; ++v) {
            const float dv = d[v];
            __builtin_nontemporal_store(
                dv, &out[(size_t)(rowbase + v) * NHID + h]);
        }
    }
}

extern "C" void kernel_launch(void* const* d_in, const int* in_sizes, int n_in,
                              void* d_out, int out_size, void* d_ws, size_t ws_size,
                              hipStream_t stream) {
    const float* x      = (const float*)d_in[0];  // (32, 8000, 1)
    const float* W_in   = (const float*)d_in[1];  // (3, 1)
    const float* b_in   = (const float*)d_in[2];  // (3,)
    const float* W_coup = (const float*)d_in[3];  // (3, 3)
    const float* W_out  = (const float*)d_in[4];  // (256, 3)
    const float* b_out  = (const float*)d_in[5];  // (256,)
    float* out = (float*)d_out;                   // (32, 8000, 256)

    float4* states = (float4*)d_ws;               // NROWS * 16 bytes = 4 MB

    // Serial part: one wave32, lane == batch.
    lorenz_scan<<<1, 32, 0, stream>>>(x, W_in, b_in, W_coup, states);

    // Streaming WMMA projection: 16000 tiles / 8 waves per block.
    lorenz_out_gemm<<<NTILES / WAVES_PER_BLOCK, 32 * WAVES_PER_BLOCK, 0, stream>>>(
        (const float*)states, W_out, b_out, out);
}